// DetectionLoss_65592740545174
// MI455X (gfx1250) — compile-verified
//
#include <hip/hip_runtime.h>
#include <hip/hip_bf16.h>
#include <math.h>

#define NB 16
#define NA 3
#define NGT 48
#define NC 80

#define FLAT0 307200   // 16*3*80*80
#define FLAT1 76800    // 16*3*40*40
#define FLAT2 19200    // 16*3*20*20
#define FLAT_TOT (FLAT0 + FLAT1 + FLAT2)

#define NCAND (3 * NB * NGT * NA)   // 6912 = 108*64
#define NPART 1664                  // 1575 blocks padded (26*64); segs 1216/320/128

#define OFF_CELLMAX 0
#define OFF_OBJT    FLAT_TOT
#define OFF_POSF    (2 * FLAT_TOT)
#define OFF_BOX     (3 * FLAT_TOT)
#define OFF_OBJPOS  (OFF_BOX + NCAND)
#define OFF_OBJNEG  (OFF_OBJPOS + NPART)
#define OFF_NPOS    (OFF_OBJNEG + NPART)
#define OFF_CLS     (OFF_NPOS + NPART)
#define TOTAL_WORDS (OFF_CLS + NPART)

typedef __attribute__((ext_vector_type(2))) float v2f;
typedef __attribute__((ext_vector_type(8))) float v8f;

#ifndef __has_builtin
#define __has_builtin(x) 0
#endif

// ---------------- numeric helpers ----------------

__device__ __forceinline__ float bce_f(float x, float t) {
  float lse = log1pf(expf(-fabsf(x)));
  float lsp = fminf(x, 0.0f) - lse;    // log_sigmoid(x)
  float lsn = fminf(-x, 0.0f) - lse;   // log_sigmoid(-x)
  return -(t * lsp + (1.0f - t) * lsn);
}

__device__ __forceinline__ float ciou_f(float p0, float p1, float p2, float p3,
                                        float g0, float g1, float g2, float g3) {
  const float eps = 1e-7f;
  float w1 = p2 - p0, h1 = p3 - p1;
  float w2 = g2 - g0, h2 = g3 - g1;
  float iw = fmaxf(fminf(p2, g2) - fmaxf(p0, g0), 0.0f);
  float ih = fmaxf(fminf(p3, g3) - fmaxf(p1, g1), 0.0f);
  float inter = iw * ih;
  float uni = w1 * h1 + w2 * h2 - inter + eps;
  float iou = inter / uni;
  float cw = fmaxf(p2, g2) - fminf(p0, g0);
  float ch = fmaxf(p3, g3) - fminf(p1, g1);
  float c2 = cw * cw + ch * ch + eps;
  float dx = g0 + g2 - p0 - p2;
  float dy = g1 + g3 - p1 - p3;
  float rho2 = (dx * dx + dy * dy) * 0.25f;
  const float k = 4.0f / (3.14159265358979323846f * 3.14159265358979323846f);
  float da = atanf(w2 / (h2 + eps)) - atanf(w1 / (h1 + eps));
  float v = k * da * da;
  float alpha = v / (v - iou + 1.0f + eps);
  return iou - (rho2 / c2 + v * alpha);
}

// GT box -> responsible grid cell
__device__ __forceinline__ void gt_cell(const float* g, float stride, int H,
                                        int& gx, int& gy,
                                        float& g0, float& g1, float& g2, float& g3) {
  g0 = g[0]; g1 = g[1]; g2 = g[2]; g3 = g[3];
  float gcx = (g0 + g2) * 0.5f;
  float gcy = (g1 + g3) * 0.5f;
  gx = min(max((int)(gcx / stride), 0), H - 1);
  gy = min(max((int)(gcy / stride), 0), H - 1);
}

// noinline so every pass computes BIT-IDENTICAL ious (win test is bit equality)
__device__ __attribute__((noinline)) void cand_iou_pos(
    const float* __restrict__ anch, int H, int gx, int gy,
    float g0, float g1, float g2, float g3,
    float* iou, bool* pos) {
  float ga = (g2 - g0) * (g3 - g1);
  for (int a = 0; a < NA; ++a) {
    const float* ab = anch + ((size_t)((a * H + gy) * H + gx)) * 4;
    float a0 = ab[0], a1 = ab[1], a2 = ab[2], a3 = ab[3];
    float w = fmaxf(fminf(g2, a2) - fmaxf(g0, a0), 0.0f);
    float h = fmaxf(fminf(g3, a3) - fmaxf(g1, a1), 0.0f);
    float inter = w * h;
    float ar = (a2 - a0) * (a3 - a1);
    iou[a] = inter / (ga + ar - inter + 1e-7f);
  }
  bool any = false;
  for (int a = 0; a < NA; ++a) { pos[a] = iou[a] > 0.5f; any = any || pos[a]; }
  if (!any) {                       // first-max fallback (matches jnp.argmax)
    int best = 0;
    if (iou[1] > iou[0]) best = 1;
    if (iou[2] > iou[best]) best = 2;
    pos[best] = true;
  }
}

// ---------------- WMMA fp32 reduction (V_WMMA_F32_16X16X4_F32, ones x B + C) ----
// n must be a multiple of 64; executed by one full wave32 (EXEC all ones).
__device__ __forceinline__ float wave_sum64(const float* __restrict__ p, int n) {
#if __has_builtin(__builtin_amdgcn_wmma_f32_16x16x4_f32)
  int lane = threadIdx.x & 31;
  v8f acc = {};
  v2f ones = {1.0f, 1.0f};
  for (int i = 0; i < n; i += 64) {
    v2f b;
    b[0] = p[i + lane];
    b[1] = p[i + 32 + lane];
    // D[m][n] = sum_k B[k][n] + C[m][n]  (A == ones): 64-wide fp32 adder tree
    acc = __builtin_amdgcn_wmma_f32_16x16x4_f32(false, ones, false, b,
                                                (short)0, acc, false, false);
  }
  float v = acc[0];                 // lanes 0-15: column sums (lanes 16-31 duplicate)
  v += __shfl_xor(v, 8);
  v += __shfl_xor(v, 4);
  v += __shfl_xor(v, 2);
  v += __shfl_xor(v, 1);
  return __shfl(v, 0);
#else
  float s = 0.0f;
  if ((threadIdx.x & 31) == 0)
    for (int i = 0; i < n; ++i) s += p[i];
  return __shfl(s, 0);
#endif
}

// ---------------- kernels ----------------

__global__ void k_init(unsigned int* __restrict__ ws, int n) {
  int i = blockIdx.x * blockDim.x + threadIdx.x;
  if (i < n) ws[i] = 0u;
}

// Pass 1: per-GT scatter-max of positive anchor IoUs into cellmax (uint bits of f32>=0)
__global__ void k_cellmax(const float* __restrict__ anch,
                          const float* __restrict__ gtb,
                          unsigned int* __restrict__ cellmax,
                          int H, float stride) {
  int tid = blockIdx.x * blockDim.x + threadIdx.x;
  if (tid >= NB * NGT) return;
  int b = tid / NGT, n = tid % NGT;
  const float* g = gtb + ((size_t)b * NGT + n) * 4;
  int gx, gy; float g0, g1, g2, g3;
  gt_cell(g, stride, H, gx, gy, g0, g1, g2, g3);
  float iou[NA]; bool pos[NA];
  cand_iou_pos(anch, H, gx, gy, g0, g1, g2, g3, iou, pos);
  for (int a = 0; a < NA; ++a)
    if (pos[a]) {
      int key = ((b * NA + a) * H + gy) * H + gx;
      atomicMax(&cellmax[key], __float_as_uint(iou[a]));
    }
}

// Pass 2: win test (bit equality), decode + CIoU, per-candidate box term,
// scatter obj_t (clipped ciou) and posf.
__global__ void k_win(const float* __restrict__ pred,
                      const float* __restrict__ anch,
                      const float* __restrict__ gtb,
                      const unsigned int* __restrict__ cellmax,
                      unsigned int* __restrict__ objt,
                      unsigned int* __restrict__ posf,
                      float* __restrict__ box_part,
                      int H, float stride) {
  int tid = blockIdx.x * blockDim.x + threadIdx.x;
  if (tid >= NB * NGT * NA) return;
  int a = tid % NA;
  int n = (tid / NA) % NGT;
  int b = tid / (NA * NGT);
  const float* g = gtb + ((size_t)b * NGT + n) * 4;
  int gx, gy; float g0, g1, g2, g3;
  gt_cell(g, stride, H, gx, gy, g0, g1, g2, g3);
  float iou[NA]; bool pos[NA];
  cand_iou_pos(anch, H, gx, gy, g0, g1, g2, g3, iou, pos);
  float boxterm = 0.0f;
  if (pos[a]) {
    int key = ((b * NA + a) * H + gy) * H + gx;
    if (__float_as_uint(iou[a]) == cellmax[key]) {
      const float* ab = anch + ((size_t)((a * H + gy) * H + gx)) * 4;
      float a0 = ab[0], a1 = ab[1], a2 = ab[2], a3 = ab[3];
      float acx = (a0 + a2) * 0.5f, acy = (a1 + a3) * 0.5f;
      float aw = a2 - a0, ah = a3 - a1;
      const float* off = pred + (size_t)key * 85;
      float s0 = 1.0f / (1.0f + expf(-off[0]));
      float s1 = 1.0f / (1.0f + expf(-off[1]));
      float s2 = 1.0f / (1.0f + expf(-off[2]));
      float s3 = 1.0f / (1.0f + expf(-off[3]));
      float cx = acx + (s0 * 2.0f - 1.0f) * stride;
      float cy = acy + (s1 * 2.0f - 1.0f) * stride;
      float w = aw * (s2 * 2.0f) * (s2 * 2.0f);
      float h = ah * (s3 * 2.0f) * (s3 * 2.0f);
      float ci = ciou_f(cx - w * 0.5f, cy - h * 0.5f, cx + w * 0.5f, cy + h * 0.5f,
                        g0, g1, g2, g3);
      boxterm = 1.0f - ci;
      float ct = fminf(fmaxf(ci, 0.0f), 1.0f);
      atomicMax(&objt[key], __float_as_uint(ct));   // nonneg -> raw-bit max == fmax
      atomicMax(&posf[key], 1u);
    }
  }
  box_part[tid] = boxterm;
}

// Pass 3: full-grid objectness BCE + class BCE at positive cells.
// Fixed-order LDS tree -> per-block partials (per-scale segments for neg/npos).
__global__ void k_grid(const float* __restrict__ pred,
                       const float* __restrict__ anch,
                       const float* __restrict__ gtb,
                       const int* __restrict__ gtl,
                       const unsigned int* __restrict__ cellmax,
                       const unsigned int* __restrict__ objt,
                       const unsigned int* __restrict__ posf,
                       float* __restrict__ objpos_part,
                       float* __restrict__ objneg_part,
                       float* __restrict__ npos_part,
                       float* __restrict__ cls_part,
                       int H, float stride, int seg_off, int glob_base) {
  int tid = threadIdx.x;
  int cell = blockIdx.x * 256 + tid;

  float x = pred[(size_t)cell * 85 + 4];
  float t = __uint_as_float(objt[cell]);
  float ol = bce_f(x, t);
  bool ispos = posf[cell] != 0u;
  float objp = ispos ? ol : 0.0f;
  float objn = ispos ? 0.0f : ol;
  float npv = ispos ? 1.0f : 0.0f;
  float clsv = 0.0f;

  if (ispos) {
    int gx = cell % H;
    int tt = cell / H;
    int gy = tt % H; tt /= H;
    int a = tt % NA;
    int b = tt / NA;
    unsigned cm = cellmax[cell];
    unsigned mask[3] = {0u, 0u, 0u};
    for (int n = 0; n < NGT; ++n) {
      const float* g = gtb + ((size_t)b * NGT + n) * 4;
      float g0, g1, g2, g3; int gxn, gyn;
      gt_cell(g, stride, H, gxn, gyn, g0, g1, g2, g3);
      if (gxn != gx || gyn != gy) continue;
      float iou[NA]; bool pos[NA];
      cand_iou_pos(anch, H, gx, gy, g0, g1, g2, g3, iou, pos);
      if (pos[a] && __float_as_uint(iou[a]) == cm) {
        int lbl = gtl[b * NGT + n];
        mask[lbl >> 5] |= (1u << (lbl & 31));
      }
    }
    const float* cl = pred + (size_t)cell * 85 + 5;
    __builtin_prefetch(cl, 0, 1);          // global_prefetch_b8
    for (int c = 0; c < NC; ++c) {
      float xc = cl[c];
      float lse = log1pf(expf(-fabsf(xc)));
      bool bit = (mask[c >> 5] >> (c & 31)) & 1u;
      float ls = (bit ? fminf(xc, 0.0f) : fminf(-xc, 0.0f)) - lse;
      clsv += -ls;
    }
  }

  __shared__ float4 sm[256];
  sm[tid] = make_float4(objp, objn, npv, clsv);
  __syncthreads();
  for (int s2 = 128; s2 > 0; s2 >>= 1) {
    if (tid < s2) {
      sm[tid].x += sm[tid + s2].x;
      sm[tid].y += sm[tid + s2].y;
      sm[tid].z += sm[tid + s2].z;
      sm[tid].w += sm[tid + s2].w;
    }
    __syncthreads();
  }
  if (tid == 0) {
    objpos_part[glob_base + blockIdx.x] = sm[0].x;
    objneg_part[seg_off + blockIdx.x]   = sm[0].y;
    npos_part[seg_off + blockIdx.x]     = sm[0].z;
    cls_part[glob_base + blockIdx.x]    = sm[0].w;
  }
}

// Pass 4: one wave32 reduces everything with WMMA and writes the 5 outputs.
__global__ void k_final(const float* __restrict__ box_part,
                        const float* __restrict__ objpos,
                        const float* __restrict__ objneg,
                        const float* __restrict__ npos,
                        const float* __restrict__ cls,
                        float* __restrict__ out) {
  float box_s  = wave_sum64(box_part, NCAND);
  float objp_s = wave_sum64(objpos, NPART);
  float cls_s  = wave_sum64(cls, NPART);
  float on0 = wave_sum64(objneg, 1216);
  float on1 = wave_sum64(objneg + 1216, 320);
  float on2 = wave_sum64(objneg + 1536, 128);
  float np0 = wave_sum64(npos, 1216);
  float np1 = wave_sum64(npos + 1216, 320);
  float np2 = wave_sum64(npos + 1536, 128);

  float npos_t = np0 + np1 + np2;
  float tp = fmaxf(npos_t, 1.0f);
  float box_loss = box_s / tp;
  float objp_loss = objp_s / tp;
  float cls_loss = cls_s / tp;
  float objn_loss = (on0 / fmaxf(307200.0f - np0, 1.0f) +
                     on1 / fmaxf(76800.0f - np1, 1.0f) +
                     on2 / fmaxf(19200.0f - np2, 1.0f)) * (1.0f / 3.0f);
  float total = box_loss + objp_loss + objn_loss + cls_loss;
  bool bad = isnan(total) || isinf(total);
  if (threadIdx.x == 0) {
    out[0] = bad ? 0.0f : total;
    out[1] = bad ? 0.0f : box_loss;
    out[2] = bad ? 0.0f : objp_loss;
    out[3] = bad ? 0.0f : objn_loss;
    out[4] = bad ? 0.0f : cls_loss;
  }
}

// ---------------- host ----------------

extern "C" void kernel_launch(void* const* d_in, const int* in_sizes, int n_in,
                              void* d_out, int out_size, void* d_ws, size_t ws_size,
                              hipStream_t stream) {
  (void)in_sizes; (void)n_in; (void)out_size; (void)ws_size;
  const float* preds[3] = {(const float*)d_in[0], (const float*)d_in[2], (const float*)d_in[4]};
  const float* anchs[3] = {(const float*)d_in[1], (const float*)d_in[3], (const float*)d_in[5]};
  const float* gtb = (const float*)d_in[6];
  const int* gtl = (const int*)d_in[7];

  unsigned int* ws = (unsigned int*)d_ws;               // ~4.9 MB scratch used
  unsigned int* cellmax = ws + OFF_CELLMAX;
  unsigned int* objt    = ws + OFF_OBJT;
  unsigned int* posf    = ws + OFF_POSF;
  float* box_part    = (float*)(ws + OFF_BOX);
  float* objpos_part = (float*)(ws + OFF_OBJPOS);
  float* objneg_part = (float*)(ws + OFF_OBJNEG);
  float* npos_part   = (float*)(ws + OFF_NPOS);
  float* cls_part    = (float*)(ws + OFF_CLS);

  static const int   Hs[3]   = {80, 40, 20};
  static const float Ss[3]   = {8.0f, 16.0f, 32.0f};
  static const int   FOFF[3] = {0, FLAT0, FLAT0 + FLAT1};
  static const int   SEG[3]  = {0, 1216, 1536};
  static const int   GB[3]   = {0, 1200, 1500};
  static const int   BLK[3]  = {1200, 300, 75};

  k_init<<<(TOTAL_WORDS + 255) / 256, 256, 0, stream>>>(ws, TOTAL_WORDS);
  for (int s = 0; s < 3; ++s)
    k_cellmax<<<3, 256, 0, stream>>>(anchs[s], gtb, cellmax + FOFF[s], Hs[s], Ss[s]);
  for (int s = 0; s < 3; ++s)
    k_win<<<9, 256, 0, stream>>>(preds[s], anchs[s], gtb, cellmax + FOFF[s],
                                 objt + FOFF[s], posf + FOFF[s],
                                 box_part + s * (NB * NGT * NA), Hs[s], Ss[s]);
  for (int s = 0; s < 3; ++s)
    k_grid<<<BLK[s], 256, 0, stream>>>(preds[s], anchs[s], gtb, gtl,
                                       cellmax + FOFF[s], objt + FOFF[s], posf + FOFF[s],
                                       objpos_part, objneg_part, npos_part, cls_part,
                                       Hs[s], Ss[s], SEG[s], GB[s]);
  k_final<<<1, 32, 0, stream>>>(box_part, objpos_part, objneg_part, npos_part, cls_part,
                                (float*)d_out);
}